// attention_pooling_69080253989330
// MI455X (gfx1250) — compile-verified
//
#include <hip/hip_runtime.h>
#include <hip/hip_bf16.h>

// Problem constants (match reference)
#define B_     64
#define N_     4096
#define D_     256     // NODE_DIM + HID_DIM
#define HID_   128
#define CHUNKS 32      // chunks per batch along N
#define ROWS   128     // rows per chunk (CHUNKS*ROWS == N_)
#define WAVES  8       // 256 threads / wave32

typedef __attribute__((ext_vector_type(2))) float v2f;
typedef __attribute__((ext_vector_type(8))) float v8f;

__device__ __forceinline__ float waveSum32(float v) {
#pragma unroll
    for (int off = 16; off > 0; off >>= 1)
        v += __shfl_xor(v, off, 32);
    return v;
}

// Kernel 1: fused scores + online softmax + weighted feature accumulation.
// One pass over feats (268 MB) -> per-(batch,chunk) partial (m, d, num[256]).
__global__ __launch_bounds__(256) void pool_partial(
    const float* __restrict__ feats, const int* __restrict__ counts,
    const float* __restrict__ Wi, const float* __restrict__ bi,
    float* __restrict__ ws_num, float* __restrict__ ws_m, float* __restrict__ ws_d)
{
    const int b     = blockIdx.y;
    const int chunk = blockIdx.x;
    const int t     = threadIdx.x;
    const int w     = t >> 5;
    const int lane  = t & 31;

    __shared__ float s_max[WAVES];
    __shared__ float s_num[D_];
    __shared__ float s_d;

    const int count = counts[b];
    int vcount = count - chunk * ROWS;
    vcount = vcount < 0 ? 0 : (vcount > ROWS ? ROWS : vcount);

    // lane-resident Wi slice: cols [4l..4l+3] and [128+4l..128+4l+3]
    const float4* wi4 = (const float4*)Wi;
    const float4 wi0 = wi4[lane];
    const float4 wi1 = wi4[32 + lane];
    const float  bi0 = bi[0];

    float4 acc0 = {0.f,0.f,0.f,0.f};
    float4 acc1 = {0.f,0.f,0.f,0.f};
    float  m_w  = -1.0e30f;   // running max (uniform per wave)
    float  d_w  = 0.0f;       // running denom (uniform per wave)

    const float4* base =
        (const float4*)(feats + ((size_t)b * N_ + (size_t)chunk * ROWS) * D_);

#pragma unroll 2
    for (int i = w; i < vcount; i += WAVES) {
        const float4* rp = base + (size_t)i * (D_ / 4);
        float4 v0 = rp[lane];        // cols 4l..4l+3      (512B coalesced)
        float4 v1 = rp[32 + lane];   // cols 128+4l..+3    (512B coalesced)

        float part = v0.x*wi0.x + v0.y*wi0.y + v0.z*wi0.z + v0.w*wi0.w
                   + v1.x*wi1.x + v1.y*wi1.y + v1.z*wi1.z + v1.w*wi1.w;
        float s = waveSum32(part) + bi0;

        float mnew = fmaxf(m_w, s);
        float corr = __expf(m_w - mnew);   // rescale old state
        float p    = __expf(s   - mnew);
        d_w = d_w * corr + p;
        acc0.x = acc0.x * corr + p * v0.x;
        acc0.y = acc0.y * corr + p * v0.y;
        acc0.z = acc0.z * corr + p * v0.z;
        acc0.w = acc0.w * corr + p * v0.w;
        acc1.x = acc1.x * corr + p * v1.x;
        acc1.y = acc1.y * corr + p * v1.y;
        acc1.z = acc1.z * corr + p * v1.z;
        acc1.w = acc1.w * corr + p * v1.w;
        m_w = mnew;
    }

    // cross-wave merge inside the block
    if (lane == 0) s_max[w] = m_w;
    s_num[t] = 0.0f;
    if (t == 0) s_d = 0.0f;
    __syncthreads();

    float m_blk = s_max[0];
#pragma unroll
    for (int j = 1; j < WAVES; ++j) m_blk = fmaxf(m_blk, s_max[j]);

    float corr = __expf(m_w - m_blk);
    atomicAdd(&s_num[4*lane + 0],       acc0.x * corr);
    atomicAdd(&s_num[4*lane + 1],       acc0.y * corr);
    atomicAdd(&s_num[4*lane + 2],       acc0.z * corr);
    atomicAdd(&s_num[4*lane + 3],       acc0.w * corr);
    atomicAdd(&s_num[128 + 4*lane + 0], acc1.x * corr);
    atomicAdd(&s_num[128 + 4*lane + 1], acc1.y * corr);
    atomicAdd(&s_num[128 + 4*lane + 2], acc1.z * corr);
    atomicAdd(&s_num[128 + 4*lane + 3], acc1.w * corr);
    if (lane == 0) atomicAdd(&s_d, d_w * corr);
    __syncthreads();

    const int pidx = b * CHUNKS + chunk;
    ws_num[(size_t)pidx * D_ + t] = s_num[t];
    if (t == 0) { ws_m[pidx] = m_blk; ws_d[pidx] = s_d; }
}

// Kernel 2: merge chunk partials per batch -> pooled[b][D_] = num/denom.
__global__ __launch_bounds__(256) void pool_combine(
    const float* __restrict__ ws_num, const float* __restrict__ ws_m,
    const float* __restrict__ ws_d, float* __restrict__ pooled)
{
    const int b = blockIdx.x;
    const int t = threadIdx.x;   // one column of D_

    float m = -1.0e30f;
#pragma unroll 4
    for (int c = 0; c < CHUNKS; ++c) m = fmaxf(m, ws_m[b * CHUNKS + c]);

    float den = 0.f, num = 0.f;
#pragma unroll 4
    for (int c = 0; c < CHUNKS; ++c) {
        float e = __expf(ws_m[b * CHUNKS + c] - m);
        den += ws_d[b * CHUNKS + c] * e;
        num += ws_num[(size_t)(b * CHUNKS + c) * D_ + t] * e;
    }
    pooled[b * D_ + t] = num / den;
}

// Kernel 3: out[64,128] = pooled[64,256] @ Wj[256,128] + bj, via
// V_WMMA_F32_16X16X4_F32. One wave per 16x16 output tile, K stepped by 4.
// A(16x4) layout: lane = {M=l&15, Khalf=2*(l>>4)}, vgpr0=K, vgpr1=K+1.
// B(4x16) layout: lane = {N=l&15, Khalf=2*(l>>4)}, vgpr0=K, vgpr1=K+1.
// C/D(16x16):     vgpr j: lanes0-15 -> M=j, lanes16-31 -> M=j+8; N=l&15.
__global__ __launch_bounds__(32) void pool_project(
    const float* __restrict__ pooled, const float* __restrict__ Wj,
    const float* __restrict__ bj, float* __restrict__ out)
{
    const int tile = blockIdx.x;     // 0..31
    const int tm   = tile >> 3;      // 0..3
    const int tn   = tile & 7;       // 0..7
    const int lane = threadIdx.x;
    const int r16  = lane & 15;
    const int half = lane >> 4;

    const int Mbase = tm * 16;
    const int Nbase = tn * 16;

    const float bjv = bj[Nbase + r16];
    v8f c;
#pragma unroll
    for (int j = 0; j < 8; ++j) c[j] = bjv;

    const float* arow = pooled + (size_t)(Mbase + r16) * D_;
#pragma unroll 4
    for (int k0 = 0; k0 < D_; k0 += 4) {
        const int k = k0 + half * 2;
        v2f a, bm;
        a[0]  = arow[k];
        a[1]  = arow[k + 1];
        bm[0] = Wj[(size_t)k       * HID_ + Nbase + r16];
        bm[1] = Wj[(size_t)(k + 1) * HID_ + Nbase + r16];
        c = __builtin_amdgcn_wmma_f32_16x16x4_f32(
                /*neg_a=*/false, a, /*neg_b=*/false, bm,
                /*c_mod=*/(short)0, c, /*reuse_a=*/false, /*reuse_b=*/false);
    }

#pragma unroll
    for (int j = 0; j < 8; ++j) {
        const int M = Mbase + j + 8 * half;
        out[(size_t)M * HID_ + Nbase + r16] = c[j];
    }
}

extern "C" void kernel_launch(void* const* d_in, const int* in_sizes, int n_in,
                              void* d_out, int out_size, void* d_ws, size_t ws_size,
                              hipStream_t stream)
{
    const float* feats  = (const float*)d_in[0];   // [64,4096,256] f32
    const int*   counts = (const int*)  d_in[1];   // [64] i32
    const float* Wi     = (const float*)d_in[2];   // [256,1] f32
    const float* bi     = (const float*)d_in[3];   // [1] f32
    const float* Wj     = (const float*)d_in[4];   // [256,128] f32
    const float* bj     = (const float*)d_in[5];   // [128] f32
    float* out = (float*)d_out;                    // [64,128] f32

    // workspace layout (floats): num | m | d | pooled  (~2.2 MB)
    float* ws      = (float*)d_ws;
    float* ws_num  = ws;                                   // 64*32*256
    float* ws_m    = ws_num + (size_t)B_ * CHUNKS * D_;    // 2048
    float* ws_d    = ws_m + B_ * CHUNKS;                   // 2048
    float* pooled  = ws_d + B_ * CHUNKS;                   // 64*256

    pool_partial<<<dim3(CHUNKS, B_), 256, 0, stream>>>(
        feats, counts, Wi, bi, ws_num, ws_m, ws_d);
    pool_combine<<<B_, 256, 0, stream>>>(ws_num, ws_m, ws_d, pooled);
    pool_project<<<(B_ / 16) * (HID_ / 16), 32, 0, stream>>>(pooled, Wj, bj, out);
}